// WeightedRuleLayer_44143673868747
// MI455X (gfx1250) — compile-verified
//
#include <hip/hip_runtime.h>
#include <cstdint>

// WeightedRuleLayer: y[n,d] = tanh( sum_k w[k,d] * layer_values[idx[n,k], d] )
// N = 500,000 neurons, K = 4, D = 64. Memory-bound random gather.
//
// CDNA5 path: per-lane random gather via GLOBAL_LOAD_ASYNC_TO_LDS_B128
// (ASYNCcnt, in-order completion), software-pipelined with ping-pong LDS
// buffers: while computing iteration i from buffer p, iteration i+1's four
// gathers are already in flight into buffer p^1 (s_wait_asynccnt 4 instead
// of 0). Indices are register-prefetched two stages ahead. Output streamed
// with non-temporal b128 stores; gather table (128 MB) is left RT-cacheable
// so it lives in the 192 MB L2.

#define KK 4
#define DD 64

typedef float f32x4 __attribute__((ext_vector_type(4)));

#if __has_builtin(__builtin_amdgcn_tanhf)
static __device__ __forceinline__ float tanh_fast(float x) {
  return __builtin_amdgcn_tanhf(x);   // gfx1250 v_tanh_f32
}
#else
static __device__ __forceinline__ float tanh_fast(float x) { return tanhf(x); }
#endif

__global__ __launch_bounds__(256) void wrl_gather_kernel(
    const float* __restrict__ lv,     // [S, 64] gather table
    const float* __restrict__ w,      // [4, 64] diagonal weights
    const int*   __restrict__ idx,    // [N*4] gather indices (int32)
    float*       __restrict__ out,    // [N, 64]
    int N)
{
  // 2 ping-pong buffers x 4 slots x float4 per thread = 128 B/thread
  // -> 32 KB per 256-thread block (WGP has 320 KB LDS; not occupancy-limiting)
  __shared__ float smem[256 * 32];

  const int tid    = threadIdx.x;
  const int lane16 = tid & 15;          // which float4 of the 64-float row
  const int dOff   = lane16 << 2;       // d offset (floats)

  // 32-bit LDS byte address of this thread's region (AS3 ptrs are LDS offsets,
  // directly usable as the VDST operand of async-to-LDS ops).
  const unsigned lds_base =
      (unsigned)(unsigned long long)
      ((__attribute__((address_space(3))) float*)(&smem[tid * 32]));

  // Hoist weights into registers: reused across all grid-stride iterations.
  f32x4 wk[KK];
#pragma unroll
  for (int k = 0; k < KK; ++k)
    wk[k] = *(const f32x4*)(w + k * DD + dOff);

  const long long totalWork = (long long)N * 16;                 // 16 lanes/neuron
  const long long stride    = (long long)gridDim.x * blockDim.x;

  long long t = (long long)blockIdx.x * blockDim.x + tid;
  if (t >= totalWork) return;

  // --- helpers -------------------------------------------------------------
  auto load_idx = [&](long long tt) -> int4 {
    // All 16 lanes of a group read the same 16 B -> one merged request.
    return *(const int4*)(idx + (size_t)(tt >> 4) * KK);
  };
  auto issue_gathers = [&](const int4& iv, unsigned bufByteOff) {
    const int iva[KK] = {iv.x, iv.y, iv.z, iv.w};
#pragma unroll
    for (int k = 0; k < KK; ++k) {
      const float*   g    = lv + (size_t)(unsigned)iva[k] * DD + dOff;
      const unsigned slot = lds_base + bufByteOff + (unsigned)(k * 16);
      asm volatile("global_load_async_to_lds_b128 %0, %1, off"
                   :
                   : "v"(slot), "v"((unsigned long long)(uintptr_t)g)
                   : "memory");
    }
  };
  // -------------------------------------------------------------------------

  // Pipeline prologue: gathers for iteration t in flight, indices for t+stride
  // already in registers.
  int4 iv = load_idx(t);
  issue_gathers(iv, 0u);
  int4 iv_next = {0, 0, 0, 0};
  if (t + stride < totalWork) iv_next = load_idx(t + stride);

  unsigned p = 0;                       // current ping-pong buffer
  for (;;) {
    const long long tn   = t + stride;
    const bool      more = tn < totalWork;

    if (more) {
      // Launch next iteration's gathers into the other buffer, then prefetch
      // indices two stages ahead. ASYNCcnt completes in order, so waiting for
      // <=4 outstanding guarantees THIS iteration's 4 transfers have landed
      // while the 4 new ones stay in flight behind the compute below.
      issue_gathers(iv_next, (p ^ 1u) * 64u);
      if (tn + stride < totalWork) iv_next = load_idx(tn + stride);
      asm volatile("s_wait_asynccnt 4" ::: "memory");
    } else {
      asm volatile("s_wait_asynccnt 0" ::: "memory");
    }

    // Consume buffer p (ds_load_b128) and do the diagonal weighted sum.
    const int n = (int)(t >> 4);
    f32x4 acc = (f32x4)0.0f;
#pragma unroll
    for (int k = 0; k < KK; ++k) {
      const f32x4 x = *(const f32x4*)(&smem[tid * 32 + p * 16 + k * 4]);
      acc += wk[k] * x;
    }

    f32x4 r;
    r.x = tanh_fast(acc.x);
    r.y = tanh_fast(acc.y);
    r.z = tanh_fast(acc.z);
    r.w = tanh_fast(acc.w);

    // Output written once, never re-read: stream it past L2 (NT store).
    __builtin_nontemporal_store(r, (f32x4*)(out + (size_t)n * DD + dOff));

    if (!more) break;
    t = tn;
    p ^= 1u;
  }
}

extern "C" void kernel_launch(void* const* d_in, const int* in_sizes, int n_in,
                              void* d_out, int out_size, void* d_ws, size_t ws_size,
                              hipStream_t stream) {
  const float* lv  = (const float*)d_in[0];   // layer_values [S,64,1] fp32
  const float* w   = (const float*)d_in[1];   // weights [4,64] fp32
  const int*   idx = (const int*)d_in[2];     // indices [N*4] int32
  float*       out = (float*)d_out;           // [N,64,1] fp32

  const int N = in_sizes[2] / KK;             // 500,000

  const int block = 256;
  const long long totalWork = (long long)N * 16;
  long long blocksNeeded = (totalWork + block - 1) / block;
  const long long blockCap = 8192;            // 2M threads: ~4 neurons/thread,
                                              // enough iterations to pipeline,
                                              // ~65K waves of memory parallelism
  const int grid = (int)(blocksNeeded < blockCap ? blocksNeeded : blockCap);

  hipLaunchKernelGGL(wrl_gather_kernel, dim3(grid), dim3(block), 0, stream,
                     lv, w, idx, out, N);
}